// TopoFPModule_11098195493236
// MI455X (gfx1250) — compile-verified
//
#include <hip/hip_runtime.h>

// ---------------------------------------------------------------- constants
#define N_SRC  8192
#define N_TGT  32768
#define C_IN   128
#define C_SKIP 64
#define C_OUT  128
#define K1     (C_IN + C_SKIP)   // 192
#define EPSF   1e-8f

typedef float v2f __attribute__((ext_vector_type(2)));
typedef float v8f __attribute__((ext_vector_type(8)));

// ================================================================ Kernel 1
// Brute-force 3-NN over 3D positions + inverse-distance weights.
// One thread per target; pos_src staged through LDS in 24KB chunks.
#define KNN_BLOCK 256
#define CHUNK     2048

__global__ __launch_bounds__(KNN_BLOCK)
void knn_kernel(const float* __restrict__ pos_src,   // [N_SRC][3]
                const float* __restrict__ pos_tgt,   // [N_TGT][3]
                int*   __restrict__ knn_idx,         // [N_TGT][3]
                float* __restrict__ knn_w)           // [N_TGT][3]
{
    __shared__ float sp[CHUNK * 3];

    const int t = blockIdx.x * KNN_BLOCK + threadIdx.x;
    const float tx = pos_tgt[t * 3 + 0];
    const float ty = pos_tgt[t * 3 + 1];
    const float tz = pos_tgt[t * 3 + 2];

    float d0 = 3.4e38f, d1 = 3.4e38f, d2 = 3.4e38f;
    int   i0 = 0,       i1 = 0,       i2 = 0;

    for (int base = 0; base < N_SRC; base += CHUNK) {
        __syncthreads();
        for (int i = threadIdx.x; i < CHUNK * 3; i += KNN_BLOCK)
            sp[i] = pos_src[base * 3 + i];
        __syncthreads();

        for (int j = 0; j < CHUNK; ++j) {
            const float dx = sp[3 * j + 0] - tx;
            const float dy = sp[3 * j + 1] - ty;
            const float dz = sp[3 * j + 2] - tz;
            const float d  = dx * dx + dy * dy + dz * dz;
            const int idx  = base + j;
            // strict '<' keeps the earliest index on ties, matching top_k
            if (d < d0)      { d2 = d1; i2 = i1; d1 = d0; i1 = i0; d0 = d; i0 = idx; }
            else if (d < d1) { d2 = d1; i2 = i1; d1 = d;  i1 = idx; }
            else if (d < d2) { d2 = d;  i2 = idx; }
        }
    }

    float w0 = 1.0f / (sqrtf(d0) + EPSF);
    float w1 = 1.0f / (sqrtf(d1) + EPSF);
    float w2 = 1.0f / (sqrtf(d2) + EPSF);
    const float inv = 1.0f / (w0 + w1 + w2);
    w0 *= inv; w1 *= inv; w2 *= inv;

    knn_idx[t * 3 + 0] = i0; knn_idx[t * 3 + 1] = i1; knn_idx[t * 3 + 2] = i2;
    knn_w  [t * 3 + 0] = w0; knn_w  [t * 3 + 1] = w1; knn_w  [t * 3 + 2] = w2;
}

// ================================================================ Kernel 2
// Fused: gather+interpolate -> concat skip -> relu(feat@W1+b1) @ W2 + b2.
// 32 targets per block; 8 wave32s; fp32 WMMA (V_WMMA_F32_16X16X4_F32).
#define MT         32                 // targets (M) per block
#define GEMM_BLOCK 256                // 8 waves
#define NWAVES     (GEMM_BLOCK / 32)
#define NTILES     ((MT / 16) * (C_OUT / 16))      // 2*8 = 16 output tiles
#define REPS       (NTILES / NWAVES)               // 2 tiles per wave (const!)
#define FS         (K1 + 1)           // 193: padded feat stride (bank-conflict free)
#define HS         (C_OUT + 1)        // 129: padded hidden stride

__global__ __launch_bounds__(GEMM_BLOCK)
void fused_mlp_kernel(const float* __restrict__ x_src,   // [N_SRC][C_IN]
                      const float* __restrict__ x_skip,  // [N_TGT][C_SKIP]
                      const int*   __restrict__ knn_idx, // [N_TGT][3]
                      const float* __restrict__ knn_w,   // [N_TGT][3]
                      const float* __restrict__ W1,      // [K1][C_OUT]
                      const float* __restrict__ b1,      // [C_OUT]
                      const float* __restrict__ W2,      // [C_OUT][C_OUT]
                      const float* __restrict__ b2,      // [C_OUT]
                      float*       __restrict__ out)     // [N_TGT][C_OUT]
{
    __shared__ float feat[MT * FS];   // 32 x 193 f32 = 24.1 KB
    __shared__ float hbuf[MT * HS];   // 32 x 129 f32 = 16.1 KB

    const int tgt0 = blockIdx.x * MT;
    const int tid  = threadIdx.x;

    // ---- stage feat tile: interpolated features (coalesced over channels)
    for (int e = tid; e < MT * C_IN; e += GEMM_BLOCK) {
        const int m = e >> 7;          // /128
        const int c = e & 127;
        const int t = tgt0 + m;
        const int j0 = knn_idx[t * 3 + 0];
        const int j1 = knn_idx[t * 3 + 1];
        const int j2 = knn_idx[t * 3 + 2];
        const float w0 = knn_w[t * 3 + 0];
        const float w1 = knn_w[t * 3 + 1];
        const float w2 = knn_w[t * 3 + 2];
        feat[m * FS + c] = w0 * x_src[j0 * C_IN + c]
                         + w1 * x_src[j1 * C_IN + c]
                         + w2 * x_src[j2 * C_IN + c];
    }
    // ---- stage skip features
    for (int e = tid; e < MT * C_SKIP; e += GEMM_BLOCK) {
        const int m = e >> 6;          // /64
        const int c = e & 63;
        feat[m * FS + C_IN + c] = x_skip[(tgt0 + m) * C_SKIP + c];
    }
    __syncthreads();

    const int lane = tid & 31;
    const int wave = tid >> 5;         // 0..7 (uniform within a wave)
    const int half = lane >> 4;        // 0: K lo pair / rows 0..7 | 1: hi
    const int l15  = lane & 15;

    // ---- layer 1: h = relu(feat @ W1 + b1)   [MT x K1] @ [K1 x C_OUT]
    // constant trip count -> fully unrolled, no divergent branch around WMMA
    #pragma unroll
    for (int rep = 0; rep < REPS; ++rep) {
        const int tile = wave + rep * NWAVES;
        const int m0 = (tile / (C_OUT / 16)) * 16;
        const int n0 = (tile % (C_OUT / 16)) * 16;
        v8f acc = {};
        #pragma unroll 4
        for (int k0 = 0; k0 < K1; k0 += 4) {
            const int ka = k0 + 2 * half;          // 16x4 fp32 A layout
            v2f a, b;
            a.x = feat[(m0 + l15) * FS + ka];
            a.y = feat[(m0 + l15) * FS + ka + 1];
            b.x = W1[(ka    ) * C_OUT + n0 + l15]; // 4x16 fp32 B layout
            b.y = W1[(ka + 1) * C_OUT + n0 + l15];
            acc = __builtin_amdgcn_wmma_f32_16x16x4_f32(
                      false, a, false, b, (short)0, acc, false, false);
        }
        const float bias = b1[n0 + l15];
        #pragma unroll
        for (int r = 0; r < 8; ++r) {
            const int row = m0 + r + 8 * half;     // C/D layout: vgpr r, half
            const float v = acc[r] + bias;
            hbuf[row * HS + n0 + l15] = v > 0.0f ? v : 0.0f;
        }
    }
    __syncthreads();

    // ---- layer 2: out = h @ W2 + b2   [MT x C_OUT] @ [C_OUT x C_OUT]
    #pragma unroll
    for (int rep = 0; rep < REPS; ++rep) {
        const int tile = wave + rep * NWAVES;
        const int m0 = (tile / (C_OUT / 16)) * 16;
        const int n0 = (tile % (C_OUT / 16)) * 16;
        v8f acc = {};
        #pragma unroll 4
        for (int k0 = 0; k0 < C_OUT; k0 += 4) {
            const int ka = k0 + 2 * half;
            v2f a, b;
            a.x = hbuf[(m0 + l15) * HS + ka];
            a.y = hbuf[(m0 + l15) * HS + ka + 1];
            b.x = W2[(ka    ) * C_OUT + n0 + l15];
            b.y = W2[(ka + 1) * C_OUT + n0 + l15];
            acc = __builtin_amdgcn_wmma_f32_16x16x4_f32(
                      false, a, false, b, (short)0, acc, false, false);
        }
        const float bias = b2[n0 + l15];
        #pragma unroll
        for (int r = 0; r < 8; ++r) {
            const int row = tgt0 + m0 + r + 8 * half;
            out[row * C_OUT + n0 + l15] = acc[r] + bias;
        }
    }
}

// ================================================================ launcher
extern "C" void kernel_launch(void* const* d_in, const int* in_sizes, int n_in,
                              void* d_out, int out_size, void* d_ws, size_t ws_size,
                              hipStream_t stream) {
    const float* x_src   = (const float*)d_in[0];
    const float* pos_src = (const float*)d_in[1];
    const float* pos_tgt = (const float*)d_in[2];
    const float* x_skip  = (const float*)d_in[3];
    const float* W1      = (const float*)d_in[4];
    const float* b1      = (const float*)d_in[5];
    const float* W2      = (const float*)d_in[6];
    const float* b2      = (const float*)d_in[7];
    float* out = (float*)d_out;

    // workspace: knn indices then knn weights
    int*   knn_idx = (int*)d_ws;
    float* knn_w   = (float*)((char*)d_ws + (size_t)N_TGT * 3 * sizeof(int));

    knn_kernel<<<N_TGT / KNN_BLOCK, KNN_BLOCK, 0, stream>>>(
        pos_src, pos_tgt, knn_idx, knn_w);

    fused_mlp_kernel<<<N_TGT / MT, GEMM_BLOCK, 0, stream>>>(
        x_src, x_skip, knn_idx, knn_w, W1, b1, W2, b2, out);
}